// TrustGeo_41532333752349
// MI455X (gfx1250) — compile-verified
//
#include <hip/hip_runtime.h>
#include <math.h>

// ---------------------------------------------------------------------------
// TrustGeo forward for MI455X (gfx1250, wave32).
// Heavy path: 5000x10000x32 attention done flash-style with
// V_WMMA_F32_16X16X4_F32 (exact f32). One wave per 16-target tile.
// ---------------------------------------------------------------------------

typedef float v2f __attribute__((ext_vector_type(2)));
typedef float v8f __attribute__((ext_vector_type(8)));

#define N1_LM 10000
#define N2_TG 5000
#define NT_LM 625   // 10000 / 16
#define NT_TG 313   // ceil(5000 / 16) -> tgQ padded to 5008 rows

// workspace layout (float offsets)
#define WS_LMK   0u        // lm_K   10000*32
#define WS_LMF   320000u   // lm_feature packed 10000*32
#define WS_TGQ   640000u   // tg_Q (pre-scaled by 1/sqrt(32)) 5008*32
#define WS_RS    800256u   // router feature sum  [32]
#define WS_DSF   800288u   // delay_score @ lm_feature [32]
#define WS_DSUM  800320u   // sum(delay_score) [1]
#define WS_R0    800352u   // router_0 [32]
#define WS_R1    800384u   // router_1 [32]

__device__ __forceinline__ v8f vz8() {
  v8f r = {0.f, 0.f, 0.f, 0.f, 0.f, 0.f, 0.f, 0.f};
  return r;
}

// D = A(16x4) * B(4x16) + C, all f32, wave32 striped per CDNA5 ISA layout.
__device__ __forceinline__ v8f wmma4(v2f a, v2f b, v8f c) {
  return __builtin_amdgcn_wmma_f32_16x16x4_f32(false, a, false, b, (short)0, c,
                                               false, false);
}

// butterfly reductions within each 16-lane half of the wave32
__device__ __forceinline__ float hmax16(float v) {
#pragma unroll
  for (int o = 1; o < 16; o <<= 1) v = fmaxf(v, __shfl_xor(v, o));
  return v;
}
__device__ __forceinline__ float hsum16(float v) {
#pragma unroll
  for (int o = 1; o < 16; o <<= 1) v += __shfl_xor(v, o);
  return v;
}

__device__ __forceinline__ float softplus_f(float x) {
  return (x > 20.0f) ? x : log1pf(__expf(x));
}

// ---------------------------------------------------------------------------
// Kernel: zero small accumulator region
// ---------------------------------------------------------------------------
__global__ void zero_kernel(float* __restrict__ p, int n) {
  int i = blockIdx.x * blockDim.x + threadIdx.x;
  if (i < n) p[i] = 0.0f;
}

// ---------------------------------------------------------------------------
// Kernel: Out[i][j] = scale * sum_{k<30} X[i][k] * W[j][k]
// One wave per 16-row tile; K padded 30->32 with zeros; pure WMMA f32.
// ---------------------------------------------------------------------------
__global__ __launch_bounds__(32) void qk_gemm_kernel(
    const float* __restrict__ X, const float* __restrict__ W,
    float* __restrict__ Out, int nrows, float scale) {
  const int lane = threadIdx.x;
  const int l16  = lane & 15;
  const int h2   = (lane >> 4) * 2;  // 0 or 2
  const int hr   = (lane >> 4) * 8;  // 0 or 8
  const int rbase = blockIdx.x * 16;

  int xr = rbase + l16;
  if (xr > nrows - 1) xr = nrows - 1;

  v2f a[8], b0[8], b1[8];
#pragma unroll
  for (int kc = 0; kc < 8; ++kc) {
    const int k = kc * 4 + h2;
    a[kc].x  = (k     < 30) ? X[xr * 30 + k]            : 0.0f;
    a[kc].y  = (k + 1 < 30) ? X[xr * 30 + k + 1]        : 0.0f;
    b0[kc].x = (k     < 30) ? W[l16 * 30 + k]           : 0.0f;
    b0[kc].y = (k + 1 < 30) ? W[l16 * 30 + k + 1]       : 0.0f;
    b1[kc].x = (k     < 30) ? W[(16 + l16) * 30 + k]    : 0.0f;
    b1[kc].y = (k + 1 < 30) ? W[(16 + l16) * 30 + k + 1] : 0.0f;
  }

  v8f d0 = vz8(), d1 = vz8();
#pragma unroll
  for (int kc = 0; kc < 8; ++kc) {
    d0 = wmma4(a[kc], b0[kc], d0);
    d1 = wmma4(a[kc], b1[kc], d1);
  }

#pragma unroll
  for (int r = 0; r < 8; ++r) {
    const int row = rbase + r + hr;  // padded Out rows are allocated
    Out[row * 32 + l16]      = scale * d0[r];
    Out[row * 32 + 16 + l16] = scale * d1[r];
  }
}

// ---------------------------------------------------------------------------
// Kernel: pack lm_feature, compute router/delay reductions
// ---------------------------------------------------------------------------
__global__ void lm_extras_kernel(
    const float* __restrict__ lm_X, const float* __restrict__ lm_Y,
    const float* __restrict__ lm_delay, const float* __restrict__ g1p,
    const float* __restrict__ ap, const float* __restrict__ bp,
    float* __restrict__ lmF, float* __restrict__ rsum,
    float* __restrict__ dsF, float* __restrict__ dsum) {
  __shared__ float srs[32];
  __shared__ float sdf[32];
  __shared__ float sds;
  const int tid = threadIdx.x;
  if (tid < 32) { srs[tid] = 0.0f; sdf[tid] = 0.0f; }
  if (tid == 0) sds = 0.0f;
  __syncthreads();

  const int i = blockIdx.x * blockDim.x + tid;
  if (i < N1_LM) {
    const float g1 = g1p[0], a = ap[0], b = bp[0];
    const float ds = __expf(-g1 * (a * lm_delay[i] + b));
    atomicAdd(&sds, ds);
#pragma unroll
    for (int j = 0; j < 32; ++j) {
      const float f = (j < 30) ? lm_X[i * 30 + j] : lm_Y[i * 2 + (j - 30)];
      lmF[i * 32 + j] = f;
      atomicAdd(&srs[j], f);
      atomicAdd(&sdf[j], ds * f);
    }
  }
  __syncthreads();
  if (tid < 32) {
    atomicAdd(&rsum[tid], srs[tid]);
    atomicAdd(&dsF[tid], sdf[tid]);
  }
  if (tid == 0) atomicAdd(dsum, sds);
}

// ---------------------------------------------------------------------------
// Kernel: router_0 / router_1  (one wave)
// ---------------------------------------------------------------------------
__global__ __launch_bounds__(32) void router_kernel(
    const float* __restrict__ rsum, const float* __restrict__ dsF,
    const float* __restrict__ dsum, const float* __restrict__ W1,
    const float* __restrict__ b1, float* __restrict__ r0out,
    float* __restrict__ r1out) {
  __shared__ float sar[32];
  const int l = threadIdx.x;
  const float r0 = rsum[l] * (1.0f / (float)N1_LM);
  const float ar = (r0 + dsF[l]) / (1.0f + dsum[0] + 1e-12f);
  sar[l] = ar;
  __syncthreads();
  float acc = b1[l];
#pragma unroll
  for (int k = 0; k < 32; ++k) acc += sar[k] * W1[l * 32 + k];
  r0out[l] = r0;
  r1out[l] = acc;  // router_1 = agg_r @ W1.T + b1
}

// ---------------------------------------------------------------------------
// (16x32)@(32x32): out = A @ W.T + bias via WMMA; A enters in D-layout,
// transposed through LDS into A-fragment layout.
// ---------------------------------------------------------------------------
__device__ __forceinline__ void tile_gemm32(
    const v8f& a0, const v8f& a1, const float* __restrict__ W,
    const float* __restrict__ bias, float* lds, int l16, int h2, int hr,
    v8f& o0, v8f& o1) {
  __syncthreads();
#pragma unroll
  for (int r = 0; r < 8; ++r) {
    lds[(r + hr) * 32 + l16]      = a0[r];
    lds[(r + hr) * 32 + 16 + l16] = a1[r];
  }
  __syncthreads();
  v8f d0 = vz8(), d1 = vz8();
#pragma unroll
  for (int kc = 0; kc < 8; ++kc) {
    const int k = kc * 4 + h2;
    v2f aa = *(const v2f*)(lds + l16 * 32 + k);
    v2f b0, b1v;
    b0.x  = W[l16 * 32 + k];
    b0.y  = W[l16 * 32 + k + 1];
    b1v.x = W[(16 + l16) * 32 + k];
    b1v.y = W[(16 + l16) * 32 + k + 1];
    d0 = wmma4(aa, b0, d0);
    d1 = wmma4(aa, b1v, d1);
  }
  const float be0 = bias[l16], be1 = bias[16 + l16];
#pragma unroll
  for (int r = 0; r < 8; ++r) {
    o0[r] = d0[r] + be0;
    o1[r] = d1[r] + be1;
  }
}

// ---------------------------------------------------------------------------
// Main attention + tail kernel: one wave handles 16 targets.
// ---------------------------------------------------------------------------
__global__ __launch_bounds__(32) void attn_kernel(
    const float* __restrict__ tgQ, const float* __restrict__ lmK,
    const float* __restrict__ lmF, const float* __restrict__ tg_X,
    const float* __restrict__ tg_delay, const float* __restrict__ router0,
    const float* __restrict__ router1, const float* __restrict__ W1,
    const float* __restrict__ b1, const float* __restrict__ W2,
    const float* __restrict__ b2, const float* __restrict__ Wg,
    const float* __restrict__ bg, const float* __restrict__ Wa,
    const float* __restrict__ ba, const float* __restrict__ g2p,
    const float* __restrict__ g3p, const float* __restrict__ ap,
    const float* __restrict__ bp, float* __restrict__ out) {
  __shared__ float lds[16 * 32];
  const int lane = threadIdx.x;
  const int l16  = lane & 15;
  const int h2   = (lane >> 4) * 2;
  const int hr   = (lane >> 4) * 8;
  const int tgbase = blockIdx.x * 16;

  // Q fragments for this 16-target tile (tgQ pre-scaled by 1/sqrt(32))
  const int qrow = tgbase + l16;  // < 5008 (padded)
  v2f aq[8];
#pragma unroll
  for (int kc = 0; kc < 8; ++kc)
    aq[kc] = *(const v2f*)(tgQ + qrow * 32 + kc * 4 + h2);

  float m_[8], Z[8], rs[8];
#pragma unroll
  for (int r = 0; r < 8; ++r) { m_[r] = -3.0e38f; Z[r] = 0.0f; rs[r] = 0.0f; }

  // ---- pass 1: online softmax statistics over all landmark tiles ----
  for (int t = 0; t < NT_LM; ++t) {
    if (t + 1 < NT_LM) __builtin_prefetch(lmK + (t + 1) * 16 * 32, 0, 1);
    const float* kb = lmK + (t * 16 + l16) * 32;
    v8f d = vz8();
#pragma unroll
    for (int kc = 0; kc < 8; ++kc)
      d = wmma4(aq[kc], *(const v2f*)(kb + kc * 4 + h2), d);
#pragma unroll
    for (int r = 0; r < 8; ++r) {
      const float mx = hmax16(d[r]);
      const float mn = fmaxf(m_[r], mx);
      const float ps = hsum16(__expf(d[r] - mn));
      Z[r] = Z[r] * __expf(m_[r] - mn) + ps;
      m_[r] = mn;
    }
  }
  float invZ[8];
#pragma unroll
  for (int r = 0; r < 8; ++r) invZ[r] = 1.0f / Z[r];

  // ---- pass 2: e = exp(softmax), accumulate e-sum and e @ lm_feature ----
  v8f acc0 = vz8(), acc1 = vz8();
  for (int t = 0; t < NT_LM; ++t) {
    if (t + 1 < NT_LM) __builtin_prefetch(lmF + (t + 1) * 16 * 32, 0, 1);
    const float* kb = lmK + (t * 16 + l16) * 32;
    v8f d = vz8();
#pragma unroll
    for (int kc = 0; kc < 8; ++kc)
      d = wmma4(aq[kc], *(const v2f*)(kb + kc * 4 + h2), d);
    float e[8];
#pragma unroll
    for (int r = 0; r < 8; ++r) {
      e[r] = __expf(__expf(d[r] - m_[r]) * invZ[r]);
      rs[r] += hsum16(e[r]);
    }
    __syncthreads();
#pragma unroll
    for (int r = 0; r < 8; ++r) lds[(r + hr) * 16 + l16] = e[r];
    __syncthreads();
#pragma unroll
    for (int c = 0; c < 4; ++c) {
      const int k0 = c * 4 + h2;
      v2f ae = *(const v2f*)(lds + l16 * 16 + k0);  // E transposed
      const float* f0 = lmF + (t * 16 + k0) * 32;
      const float* f1 = lmF + (t * 16 + k0 + 1) * 32;
      v2f bf0, bf1;
      bf0.x = f0[l16];      bf0.y = f1[l16];
      bf1.x = f0[16 + l16]; bf1.y = f1[16 + l16];
      acc0 = wmma4(ae, bf0, acc0);
      acc1 = wmma4(ae, bf1, acc1);
    }
  }

  // ---- per-target tail ----
  const float g2 = g2p[0], g3 = g3p[0], av = ap[0], bv = bp[0];
  float td8[8], x0[8], x1[8];
  v8f agg0, agg1;
#pragma unroll
  for (int r = 0; r < 8; ++r) {
    int gm = tgbase + r + hr;
    const int gmc = (gm < N2_TG) ? gm : (N2_TG - 1);
    const float td = tg_delay[gmc];
    td8[r] = td;
    const float r20 = __expf(-g2 * (av * td + bv));
    x0[r] = (l16 < 30)      ? tg_X[gmc * 30 + l16]      : 0.0f;
    x1[r] = (16 + l16 < 30) ? tg_X[gmc * 30 + 16 + l16] : 0.0f;
    const float deg = 1.0f + rs[r] + r20 + 1e-12f;
    agg0[r] = (x0[r] + acc0[r] + r20 * router0[l16]) / deg;
    agg1[r] = (x1[r] + acc1[r] + r20 * router0[16 + l16]) / deg;
  }

  v8f tgf1_0, tgf1_1;
  tile_gemm32(agg0, agg1, W1, b1, lds, l16, h2, hr, tgf1_0, tgf1_1);

  v8f a1_0, a1_1;
#pragma unroll
  for (int r = 0; r < 8; ++r) {
    const float r21 = __expf(-g3 * (av * td8[r] + bv));
    const float den = 1.0f + r21 + 1e-12f;
    a1_0[r] = (tgf1_0[r] + r21 * router1[l16]) / den;
    a1_1[r] = (tgf1_1[r] + r21 * router1[16 + l16]) / den;
  }

  v8f tgf2_0, tgf2_1;
  tile_gemm32(a1_0, a1_1, W2, b2, lds, l16, h2, hr, tgf2_0, tgf2_1);

  // heads
#pragma unroll
  for (int r = 0; r < 8; ++r) {
    const int gm = tgbase + r + hr;
    float og[5], oa[5];
#pragma unroll
    for (int c = 0; c < 5; ++c) {
      float pg = tgf1_0[r] * Wg[c * 64 + l16] +
                 tgf1_1[r] * Wg[c * 64 + 16 + l16] +
                 tgf2_0[r] * Wg[c * 64 + 32 + l16] +
                 tgf2_1[r] * Wg[c * 64 + 48 + l16];
      og[c] = hsum16(pg) + bg[c];
      const float wa1 = (16 + l16 < 30) ? Wa[c * 30 + 16 + l16] : 0.0f;
      float pa = x0[r] * Wa[c * 30 + l16] + x1[r] * wa1;
      oa[c] = hsum16(pa) + ba[c];
    }
    if (gm < N2_TG) {
      if (l16 == 0)      out[2 * gm]              = og[0];
      else if (l16 == 1) out[2 * gm + 1]          = og[1];
      else if (l16 == 2) out[10000 + gm]          = softplus_f(og[2]);
      else if (l16 == 3) out[15000 + gm]          = softplus_f(og[3]) + 1.0f;
      else if (l16 == 4) out[20000 + gm]          = softplus_f(og[4]);
      else if (l16 == 5) out[25000 + 2 * gm]      = oa[0];
      else if (l16 == 6) out[25000 + 2 * gm + 1]  = oa[1];
      else if (l16 == 7) out[35000 + gm]          = softplus_f(oa[2]);
      else if (l16 == 8) out[40000 + gm]          = softplus_f(oa[3]) + 1.0f;
      else if (l16 == 9) out[45000 + gm]          = softplus_f(oa[4]);
    }
  }
}

// ---------------------------------------------------------------------------
extern "C" void kernel_launch(void* const* d_in, const int* in_sizes, int n_in,
                              void* d_out, int out_size, void* d_ws,
                              size_t ws_size, hipStream_t stream) {
  (void)in_sizes; (void)n_in; (void)out_size; (void)ws_size;
  const float* lm_X     = (const float*)d_in[0];
  const float* lm_Y     = (const float*)d_in[1];
  const float* tg_X     = (const float*)d_in[2];
  const float* lm_delay = (const float*)d_in[4];
  const float* tg_delay = (const float*)d_in[5];
  const float* Wq       = (const float*)d_in[6];
  const float* Wk       = (const float*)d_in[7];
  const float* g1       = (const float*)d_in[9];
  const float* g2       = (const float*)d_in[10];
  const float* g3       = (const float*)d_in[11];
  const float* al       = (const float*)d_in[12];
  const float* be       = (const float*)d_in[13];
  const float* W1       = (const float*)d_in[14];
  const float* b1       = (const float*)d_in[15];
  const float* W2       = (const float*)d_in[16];
  const float* b2       = (const float*)d_in[17];
  const float* Wg       = (const float*)d_in[18];
  const float* bg       = (const float*)d_in[19];
  const float* Wa       = (const float*)d_in[20];
  const float* ba       = (const float*)d_in[21];
  float* ws  = (float*)d_ws;
  float* out = (float*)d_out;

  const float invtemp = 1.0f / sqrtf(32.0f);

  zero_kernel<<<1, 128, 0, stream>>>(ws + WS_RS, 96);
  qk_gemm_kernel<<<NT_LM, 32, 0, stream>>>(lm_X, Wk, ws + WS_LMK, N1_LM, 1.0f);
  qk_gemm_kernel<<<NT_TG, 32, 0, stream>>>(tg_X, Wq, ws + WS_TGQ, N2_TG, invtemp);
  lm_extras_kernel<<<40, 256, 0, stream>>>(lm_X, lm_Y, lm_delay, g1, al, be,
                                           ws + WS_LMF, ws + WS_RS, ws + WS_DSF,
                                           ws + WS_DSUM);
  router_kernel<<<1, 32, 0, stream>>>(ws + WS_RS, ws + WS_DSF, ws + WS_DSUM,
                                      W1, b1, ws + WS_R0, ws + WS_R1);
  attn_kernel<<<NT_TG, 32, 0, stream>>>(
      ws + WS_TGQ, ws + WS_LMK, ws + WS_LMF, tg_X, tg_delay, ws + WS_R0,
      ws + WS_R1, W1, b1, W2, b2, Wg, bg, Wa, ba, g2, g3, al, be, out);
}